// BetterMCMC_14714557956384
// MI455X (gfx1250) — compile-verified
//
#include <hip/hip_runtime.h>
#include <hip/hip_bf16.h>
#include <stdint.h>

// ---------------------------------------------------------------------------
// MH permutation chain for MI455X (gfx1250).
// Single persistent 1024-thread workgroup (32 wave32 waves), chain state in
// LDS, TDM strided column DMA + row DMA into LDS, Gumbel-max categorical
// (softmax cancels in the MH ratio), O(1) delta chain-weight update,
// WMMA f32 16x16x4 reduction for the initial chain weight.
// ---------------------------------------------------------------------------

#define N_WORDS   4096
#define N_SAMPLES 256
#define N_THIN    10
#define CHAIN     (N_SAMPLES * N_THIN)   // 2560 states, 2559 steps
#define NTHREADS  1024

typedef __attribute__((ext_vector_type(2))) float        v2f;
typedef __attribute__((ext_vector_type(8))) float        v8f;
typedef __attribute__((ext_vector_type(4))) unsigned int u32x4;
typedef __attribute__((ext_vector_type(4))) int          i32x4;
typedef __attribute__((ext_vector_type(8))) int          i32x8;

// ----------------------------- RNG (threefry2x32) ---------------------------
__device__ __forceinline__ unsigned rotl32(unsigned x, int r) {
  return (x << r) | (x >> (32 - r));
}

__device__ __forceinline__ void threefry2x32(unsigned k0, unsigned k1,
                                             unsigned c0, unsigned c1,
                                             unsigned& o0, unsigned& o1) {
  const unsigned ks[3] = {k0, k1, k0 ^ k1 ^ 0x1BD11BDAu};
  const int R[8] = {13, 15, 26, 6, 17, 29, 16, 24};
  unsigned x0 = c0 + ks[0], x1 = c1 + ks[1];
#pragma unroll
  for (int j = 0; j < 5; ++j) {
#pragma unroll
    for (int r = 0; r < 4; ++r) {
      x0 += x1;
      x1 = rotl32(x1, R[(j & 1) * 4 + r]);
      x1 ^= x0;
    }
    x0 += ks[(j + 1) % 3];
    x1 += ks[(j + 2) % 3] + (unsigned)(j + 1);
  }
  o0 = x0; o1 = x1;
}

__device__ __forceinline__ float u01(unsigned b) {
  // strictly inside (0,1) so -log(-log(u)) is finite
  return ((float)(b >> 8) + 0.5f) * (1.0f / 16777216.0f);
}

// ----------------------- Tensor Data Mover descriptor -----------------------
// 2D tile DMA: global (strided) -> LDS (contiguous). data_size = 4 bytes.
// clang-23 toolchain: 6-arg builtin (g0, g1, g2, g3, g4, cpol).
__device__ __forceinline__ void tdm_load_2d(unsigned lds_addr,
                                            unsigned long long gaddr,
                                            unsigned tile0, unsigned tile1,
                                            unsigned td0, unsigned td1,
                                            unsigned long long stride0) {
  u32x4 g0;
  g0[0] = 1u;                                           // count=1, user D#
  g0[1] = lds_addr;                                     // LDS byte address
  g0[2] = (unsigned)(gaddr & 0xFFFFFFFFull);            // global_addr[31:0]
  g0[3] = (unsigned)((gaddr >> 32) & 0x1FFFFFFull)      // global_addr[56:32]
          | (2u << 30);                                 // type = 2 ("image")
  i32x8 g1;
  g1[0] = (int)(2u << 16);                              // data_size=2 -> 4B
  g1[1] = (int)((td0 & 0xFFFFu) << 16);                 // tensor_dim0[15:0]
  g1[2] = (int)(((td0 >> 16) & 0xFFFFu) | ((td1 & 0xFFFFu) << 16));
  g1[3] = (int)(((td1 >> 16) & 0xFFFFu) | ((tile0 & 0xFFFFu) << 16));
  g1[4] = (int)(tile1 & 0xFFFFu);                       // tile_dim1 (tile_dim2=0)
  g1[5] = (int)(unsigned)(stride0 & 0xFFFFFFFFull);     // tensor_dim0_stride lo
  g1[6] = (int)(unsigned)((stride0 >> 32) & 0xFFFFull); // stride hi (dim1_stride=0)
  g1[7] = 0;
  i32x4 z4 = {0, 0, 0, 0};
  i32x8 z8 = {0, 0, 0, 0, 0, 0, 0, 0};
  __builtin_amdgcn_tensor_load_to_lds(g0, g1, z4, z4, z8, 0);
}

// ------------------------------ LDS state -----------------------------------
__shared__ int      s_permA[N_WORDS];
__shared__ int      s_permB[N_WORDS];
__shared__ float    s_col[N_WORDS];   // bigram[:, to_move] (also w0 scratch)
__shared__ float    s_row[N_WORDS];   // bigram[to_move, :]
__shared__ float    s_rv[32];
__shared__ int      s_ri[32];
__shared__ int      s_pos, s_tm, s_np, s_acc;
__shared__ unsigned s_kc0, s_kc1;
__shared__ float    s_uacc, s_w, s_stm, s_etm;

// insertion logit for slot i of perm2 (perm with element at `pos` removed)
__device__ __forceinline__ float logit_at(int i, int pos, const int* cur,
                                          float stm, float etm) {
  float l = 0.f;
  if (i == 0) {
    l += stm;
  } else {
    int j = i - 1;
    l += s_col[cur[j + (j >= pos)]];
  }
  if (i == N_WORDS - 1) {
    l += etm;
  } else {
    l += s_row[cur[i + (i >= pos)]];
  }
  return l;
}

__global__ void __launch_bounds__(NTHREADS)
mh_chain_kernel(const float* __restrict__ bigram,
                const float* __restrict__ start,
                const float* __restrict__ endv,
                int* __restrict__ out,
                float* __restrict__ ws) {
  const int tid  = threadIdx.x;
  const int lane = tid & 31;
  const int wid  = __builtin_amdgcn_readfirstlane((int)(threadIdx.x >> 5));

  // ---- init: identity permutation + per-thread partial chain weight -------
  for (int i = tid; i < N_WORDS; i += NTHREADS) s_permA[i] = i;
  float part = 0.f;
  for (int k = tid; k < N_WORDS - 1; k += NTHREADS)
    part += bigram[(size_t)k * N_WORDS + k + 1];
  s_col[tid] = part;  // 1024 partials, reduced on the matrix pipe below
  __syncthreads();

  // ---- w0 reduction via V_WMMA_F32_16X16X4_F32 (wave 0, EXEC all-ones) ----
  if (wid == 0) {
    v8f acc = {};
#pragma unroll
    for (int c = 0; c < 16; ++c) {
      v2f a; a[0] = s_col[c * 64 + lane * 2]; a[1] = s_col[c * 64 + lane * 2 + 1];
      v2f b = {1.0f, 1.0f};
      // D[m][n] = sum_k A[m][k]*1 + C[m][n] : accumulates row-sums of partials
      acc = __builtin_amdgcn_wmma_f32_16x16x4_f32(
          false, a, false, b, (short)0, acc, false, false);
    }
    float t = acc[0] + acc[1] + acc[2] + acc[3] + acc[4] + acc[5] + acc[6] + acc[7];
    t += __shfl_xor(t, 16);  // combine M=0..7 half with M=8..15 half
    if (lane == 0) s_w = t + start[0] + endv[N_WORDS - 1];
  }
  __syncthreads();

  // ------------------------------ the chain --------------------------------
  for (int t = 1; t < CHAIN; ++t) {
    int* cur = (t & 1) ? s_permA : s_permB;
    int* nxt = (t & 1) ? s_permB : s_permA;

    // (1) per-step randomness (thread 0), key(0,42) style threefry tree
    if (tid == 0) {
      unsigned st0, st1, a0, a1, b0, b1, c0, c1;
      threefry2x32(0u, 42u, (unsigned)t, 0u, st0, st1);
      threefry2x32(st0, st1, 0u, 0u, a0, a1);  // position key
      threefry2x32(st0, st1, 1u, 0u, b0, b1);  // categorical key
      threefry2x32(st0, st1, 2u, 0u, c0, c1);  // accept key
      int pos = (int)(a0 & (N_WORDS - 1));
      s_pos  = pos;
      s_tm   = cur[pos];
      s_kc0  = b0; s_kc1 = b1;
      s_uacc = u01(c0);
      s_stm  = start[s_tm];
      s_etm  = endv[s_tm];
    }
    __syncthreads();

    const int   pos = s_pos;
    const int   tm  = s_tm;
    const float stm = s_stm;
    const float etm = s_etm;

    // (2) TDM: DMA the column (stride 4096) and row (contiguous) into LDS
    if (wid == 0) {
      unsigned utm = (unsigned)__builtin_amdgcn_readfirstlane(tm);
      unsigned long long base = (unsigned long long)(uintptr_t)bigram;
      tdm_load_2d((unsigned)(uintptr_t)(void*)s_col, base + 4ull * utm,
                  1u, N_WORDS, N_WORDS, N_WORDS, (unsigned long long)N_WORDS);
      tdm_load_2d((unsigned)(uintptr_t)(void*)s_row,
                  base + 4ull * utm * N_WORDS,
                  N_WORDS, 1u, N_WORDS, 1u, (unsigned long long)N_WORDS);
      __builtin_amdgcn_s_wait_tensorcnt(0);
    }
    __syncthreads();

    // (3) Gumbel-max categorical: argmax(logit + gumbel) over 4096 slots
    float bestV = -__builtin_huge_valf();
    int   bestI = 0;
#pragma unroll
    for (int q = 0; q < 4; ++q) {
      int i = tid + q * NTHREADS;  // stride-1024 -> conflict-free LDS access
      float l = logit_at(i, pos, cur, stm, etm);
      unsigned g0, g1;
      threefry2x32(s_kc0, s_kc1, 0x67756D62u, (unsigned)i, g0, g1);
      float v = l - __logf(-__logf(u01(g0)));
      if (v > bestV || (v == bestV && i < bestI)) { bestV = v; bestI = i; }
    }
#pragma unroll
    for (int off = 16; off; off >>= 1) {
      float ov = __shfl_down(bestV, off);
      int   oi = __shfl_down(bestI, off);
      if (ov > bestV || (ov == bestV && oi < bestI)) { bestV = ov; bestI = oi; }
    }
    if (lane == 0) { s_rv[wid] = bestV; s_ri[wid] = bestI; }
    __syncthreads();

    // (4) accept/reject via O(1) delta weight (thread 0)
    if (tid == 0) {
      float bv = s_rv[0]; int bi = s_ri[0];
      for (int w = 1; w < 32; ++w)
        if (s_rv[w] > bv || (s_rv[w] == bv && s_ri[w] < bi)) { bv = s_rv[w]; bi = s_ri[w]; }
      int np = bi;

      float lnp  = logit_at(np,  pos, cur, stm, etm);
      float lpos = logit_at(pos, pos, cur, stm, etm);

      // removal: w(perm2) = w - A + B
      float A, B;
      if (pos == 0) {
        A = stm + s_row[cur[1]];
        B = start[cur[1]];
      } else if (pos == N_WORDS - 1) {
        A = etm + s_col[cur[N_WORDS - 2]];
        B = endv[cur[N_WORDS - 2]];
      } else {
        A = s_col[cur[pos - 1]] + s_row[cur[pos + 1]];
        B = bigram[(size_t)cur[pos - 1] * N_WORDS + cur[pos + 1]];
      }
      // insertion: w(sample) = w(perm2) + lnp - C
      float C;
      if (np == 0) {
        C = start[cur[0 + (0 >= pos)]];
      } else if (np == N_WORDS - 1) {
        int j = N_WORDS - 2;
        C = endv[cur[j + (j >= pos)]];
      } else {
        int ja = np - 1, jb = np;
        C = bigram[(size_t)cur[ja + (ja >= pos)] * N_WORDS + cur[jb + (jb >= pos)]];
      }
      float dW  = lnp - C - A + B;                 // new_w - w_last
      float acc = fminf(1.0f, __expf(dW + lnp - lpos));  // softmax cancels
      int ok = (acc > s_uacc) ? 1 : 0;
      s_np  = np;
      s_acc = ok;
      if (ok) s_w += dW;                           // keep tracked weight live
    }
    __syncthreads();

    // (5) materialize next state (+ thinned emission every 10th step)
    const int  np   = s_np;
    const int  ok   = s_acc;
    const bool emit = ((t % N_THIN) == (N_THIN - 1));
    int* orow = out + (size_t)(t / N_THIN) * N_WORDS;
#pragma unroll
    for (int q = 0; q < 4; ++q) {
      int i = tid + q * NTHREADS;
      int v;
      if (!ok) {
        v = cur[i];
      } else if (i == np) {
        v = tm;
      } else {
        int j2 = i - (i > np);
        v = cur[j2 + (j2 >= pos)];
      }
      nxt[i] = v;
      if (emit) orow[i] = v;
    }
    __syncthreads();
  }

  if (tid == 0) ws[0] = s_w;  // final chain weight -> scratch (keeps w0 live)
}

extern "C" void kernel_launch(void* const* d_in, const int* in_sizes, int n_in,
                              void* d_out, int out_size, void* d_ws, size_t ws_size,
                              hipStream_t stream) {
  // inputs: [0] n_words (scalar int), [1] bigram f32[4096*4096],
  //         [2] start f32[4096], [3] end f32[4096]
  const float* bigram = (const float*)d_in[1];
  const float* start  = (const float*)d_in[2];
  const float* endv   = (const float*)d_in[3];
  int*   out = (int*)d_out;
  float* ws  = (float*)d_ws;
  hipLaunchKernelGGL(mh_chain_kernel, dim3(1), dim3(NTHREADS), 0, stream,
                     bigram, start, endv, out, ws);
}